// Conv2dApproximation_44676249813637
// MI455X (gfx1250) — compile-verified
//
#include <hip/hip_runtime.h>

typedef __attribute__((ext_vector_type(2))) float v2f;
typedef __attribute__((ext_vector_type(8))) float v8f;

#define CIN    256
#define HH     64
#define WW     64
#define COUT   256
#define OHH    62
#define OWW    62
#define NBATCH 16
#define KTOT   (CIN * 9)             // 2304
#define NTOT   (NBATCH * OHH * OWW)  // 61504 = 961 * 64 exactly
#define BM     128
#define BN     64
#define BK     32
#define LDA    36                    // padded LDS row stride in dwords

__global__ __launch_bounds__(256, 2)
void conv2d_implicit_gemm_wmma_f32(const float* __restrict__ x,
                                   const float* __restrict__ wgt,
                                   const float* __restrict__ bias,
                                   float* __restrict__ out)
{
    __shared__ float Alds[BM * LDA];   // [m_local][k_local], stride 36
    __shared__ float Blds[BN * LDA];   // [n_local][k_local], stride 36

    const int tid  = threadIdx.x;
    const int lane = tid & 31;
    const int wv   = tid >> 5;         // wave id 0..7 (wave32)

    const int blockN = blockIdx.x;     // 0..960 (N tiles of 64)
    const int blockM = blockIdx.y;     // 0..1   (M tiles of 128)

    // ---- A (weights, OIHW flat = [COUT][KTOT]) staging:
    //      thread -> m_local = tid>>1 (0..127), k start = (tid&1)*16 (16 floats)
    const int aM  = tid >> 1;
    const int aKs = (tid & 1) * 16;
    const float* Ag = wgt + (size_t)(blockM * BM + aM) * KTOT + aKs;

    // ---- B (im2col of x) staging:
    //      thread -> n_local = tid&63 (coalesced over lanes), k group = (tid>>6)*8
    const int bN  = tid & 63;
    const int bKg = (tid >> 6) * 8;    // 0,8,16,24
    const int n   = blockN * BN + bN;  // always < NTOT (exact tiling)
    const int nb  = n / (OHH * OWW);
    const int nr  = n - nb * (OHH * OWW);
    const int oh  = nr / OWW;
    const int ow  = nr - oh * OWW;
    const int xbase = nb * CIN * (HH * WW) + oh * WW + ow; // + c*4096 + kh*64 + kw

    float4 aR0, aR1, aR2, aR3;
    float  bR[8];

    auto loadA = [&](int kc) {
        aR0 = *(const float4*)(Ag + kc);
        aR1 = *(const float4*)(Ag + kc + 4);
        aR2 = *(const float4*)(Ag + kc + 8);
        aR3 = *(const float4*)(Ag + kc + 12);
    };
    auto loadB = [&](int kc) {
#pragma unroll
        for (int i = 0; i < 8; ++i) {
            int k  = kc + bKg + i;
            int c  = k / 9;
            int rr = k - c * 9;
            int kh = rr / 3;
            int kw = rr - kh * 3;
            bR[i] = x[xbase + c * (HH * WW) + kh * WW + kw];
        }
    };
    auto stageLDS = [&]() {
        *(float4*)&Alds[aM * LDA + aKs]      = aR0;  // 16B aligned: 36%4==0, aKs%16==0
        *(float4*)&Alds[aM * LDA + aKs + 4]  = aR1;
        *(float4*)&Alds[aM * LDA + aKs + 8]  = aR2;
        *(float4*)&Alds[aM * LDA + aKs + 12] = aR3;
        float4 b0 = make_float4(bR[0], bR[1], bR[2], bR[3]);
        float4 b1 = make_float4(bR[4], bR[5], bR[6], bR[7]);
        *(float4*)&Blds[bN * LDA + bKg]     = b0;
        *(float4*)&Blds[bN * LDA + bKg + 4] = b1;
    };

    // ---- wave tile mapping: 8 waves, wave wv owns rows [wv*16, wv*16+16) x all 64 N cols
    const int laneM  = lane & 15;
    const int laneHi = lane >> 4;      // 0/1: ISA A/B layout -> K += 2 for lanes 16..31

    const int aRow = (wv * 16 + laneM) * LDA;
    const int bC0  = ( 0 + laneM) * LDA;
    const int bC1  = (16 + laneM) * LDA;
    const int bC2  = (32 + laneM) * LDA;
    const int bC3  = (48 + laneM) * LDA;

    v8f acc0 = {}, acc1 = {}, acc2 = {}, acc3 = {};

    // software pipeline: stage chunk kc from regs, prefetch kc+BK, compute kc
    loadA(0);
    loadB(0);

    for (int kc = 0; kc < KTOT; kc += BK) {
        __syncthreads();               // previous chunk's WMMA reads done
        stageLDS();
        __syncthreads();               // LDS tile visible to all waves
        if (kc + BK < KTOT) { loadA(kc + BK); loadB(kc + BK); }

#pragma unroll
        for (int ks = 0; ks < BK; ks += 4) {
            const int kb = ks + laneHi * 2; // lanes 0-15: K={ks,ks+1}; 16-31: {ks+2,ks+3}
            v2f a  = *(const v2f*)&Alds[aRow + kb];
            v2f b0 = *(const v2f*)&Blds[bC0  + kb];
            v2f b1 = *(const v2f*)&Blds[bC1  + kb];
            v2f b2 = *(const v2f*)&Blds[bC2  + kb];
            v2f b3 = *(const v2f*)&Blds[bC3  + kb];
            acc0 = __builtin_amdgcn_wmma_f32_16x16x4_f32(
                       false, a, false, b0, (short)0, acc0, false, false);
            acc1 = __builtin_amdgcn_wmma_f32_16x16x4_f32(
                       false, a, false, b1, (short)0, acc1, false, false);
            acc2 = __builtin_amdgcn_wmma_f32_16x16x4_f32(
                       false, a, false, b2, (short)0, acc2, false, false);
            acc3 = __builtin_amdgcn_wmma_f32_16x16x4_f32(
                       false, a, false, b3, (short)0, acc3, false, false);
        }
    }

    // ---- epilogue: C/D layout: VGPR v, lanes 0-15 -> M = v; lanes 16-31 -> M = v+8
    const int mg = blockM * BM + wv * 16 + laneHi * 8;

    v8f accs[4] = {acc0, acc1, acc2, acc3};
#pragma unroll
    for (int j = 0; j < 4; ++j) {
        const int nj = blockN * BN + j * 16 + laneM;
        const int bj = nj / (OHH * OWW);
        const int rj = nj - bj * (OHH * OWW);
        float* op = out + (size_t)(bj * COUT + mg) * (OHH * OWW) + rj;
#pragma unroll
        for (int v = 0; v < 8; ++v)
            op[(size_t)v * (OHH * OWW)] = accs[j][v] + bias[mg + v];
    }
}

extern "C" void kernel_launch(void* const* d_in, const int* in_sizes, int n_in,
                              void* d_out, int out_size, void* d_ws, size_t ws_size,
                              hipStream_t stream) {
    (void)in_sizes; (void)n_in; (void)out_size; (void)d_ws; (void)ws_size;
    const float* x    = (const float*)d_in[0];
    const float* wgt  = (const float*)d_in[1];
    const float* bias = (const float*)d_in[2];
    float* out        = (float*)d_out;

    dim3 grid(NTOT / BN, COUT / BM);   // 961 x 2 workgroups
    conv2d_implicit_gemm_wmma_f32<<<grid, dim3(256), 0, stream>>>(x, wgt, bias, out);
}